// AnchorTargetLayer_18640158065288
// MI455X (gfx1250) — compile-verified
//
#include <hip/hip_runtime.h>
#include <hip/hip_bf16.h>
#include <stdint.h>

#define B_    16
#define A_    9
#define H_    64
#define W_    64
#define M_    50
#define T_    (H_*W_*A_)        // 36864 == 144*256 (all waves fully active)
#define NUMFG 128
#define RPNB  256
#define NBIN  4096
#define CAP_  1024

typedef unsigned int v4u __attribute__((ext_vector_type(4)));
typedef int          v8i __attribute__((ext_vector_type(8)));
typedef int          v4i __attribute__((ext_vector_type(4)));

// ---------- order-preserving float<->int key (handles negatives) ----------
__device__ __forceinline__ int fkey(float f) {
    int k = __float_as_int(f);
    return k >= 0 ? k : (k ^ 0x7fffffff);
}
__device__ __forceinline__ float kinv(int k) {
    return __int_as_float(k >= 0 ? k : (k ^ 0x7fffffff));
}

// ---------- bit-exact IoU used by both passes ----------
__device__ __forceinline__ float iou_one(float ax1, float ay1, float ax2, float ay2,
                                         float areaA,
                                         float gx1, float gy1, float gx2, float gy2) {
    float iw = fminf(ax2, gx2) - fmaxf(ax1, gx1) + 1.0f; iw = fmaxf(iw, 0.0f);
    float ih = fminf(ay2, gy2) - fmaxf(ay1, gy1) + 1.0f; ih = fmaxf(ih, 0.0f);
    float areaG = (gx2 - gx1 + 1.0f) * (gy2 - gy1 + 1.0f);
    float inter = iw * ih;
    return inter / (areaA + areaG - inter);
}

// ---------- TDM: DMA gt_boxes[b] (250 dwords) into LDS via tensor_load_to_lds ----
// Called by threads tid<32 only (wave 0); other waves branch around it.
// Wave 0 issues once, waits on TENSORcnt; caller's __syncthreads() publishes.
__device__ __forceinline__ void tdm_load_dwords(const float* __restrict__ gsrc,
                                                float* lds_dst, int ndw, int tid) {
#if defined(__HIP_DEVICE_COMPILE__) && defined(__gfx1250__)
    unsigned lds_off = (unsigned)(unsigned long long)(void*)lds_dst; // low 32b of flat LDS ptr = LDS byte offset
    unsigned long long ga = (unsigned long long)gsrc;
    unsigned d0 = (unsigned)ndw;

    v4u g0;
    g0.x = 1u;                                               // count=1, user mode, no gather
    g0.y = lds_off;                                          // lds_addr (bytes)
    g0.z = (unsigned)(ga & 0xffffffffull);                   // global_addr[31:0]
    g0.w = (unsigned)((ga >> 32) & 0x1ffffffull) | 0x80000000u; // global_addr[56:32] | type=2

    v8i g1;
    g1[0] = 0x00020000;                        // workgroup_mask=0, data_size=2 (4B)
    g1[1] = (int)(d0 << 16);                   // tensor_dim0[15:0] @bits[63:48]
    g1[2] = (int)((d0 >> 16) | (1u << 16));    // tensor_dim0[31:16] | tensor_dim1=1
    g1[3] = (int)(d0 << 16);                   // tensor_dim1 hi=0 | tile_dim0 = ndw
    g1[4] = 1;                                 // tile_dim1=1, tile_dim2=0
    g1[5] = (int)d0;                           // tensor_dim0_stride[31:0]
    g1[6] = 0;                                 // stride hi | tensor_dim1_stride lo
    g1[7] = 0;

    v4i z4 = {0, 0, 0, 0};
#if __clang_major__ >= 23
    v8i z8 = {0, 0, 0, 0, 0, 0, 0, 0};
    __builtin_amdgcn_tensor_load_to_lds(g0, g1, z4, z4, z8, 0);
#else
    __builtin_amdgcn_tensor_load_to_lds(g0, g1, z4, z4, 0);
#endif
    __builtin_amdgcn_s_wait_tensorcnt(0);
#else
    for (int i = tid; i < ndw; i += 32) lds_dst[i] = gsrc[i];
#endif
}

// ---------- anchor reconstruction (t = (y*W+x)*A + a) ----------
__device__ __forceinline__ void anchor_of(const float* __restrict__ anchors, int t,
                                          float& ax1, float& ay1, float& ax2, float& ay2,
                                          int& xi, int& yi, int& ai) {
    int s = t / A_;
    ai = t - s * A_;
    xi = s % W_;
    yi = s / W_;
    float sx = (float)xi * 16.0f, sy = (float)yi * 16.0f;
    ax1 = anchors[ai * 4 + 0] + sx;
    ay1 = anchors[ai * 4 + 1] + sy;
    ax2 = anchors[ai * 4 + 2] + sx;
    ay2 = anchors[ai * 4 + 3] + sy;
}

// ================= kernels =================

__global__ void k_init(int* __restrict__ gtk, int* __restrict__ sumfg, int* __restrict__ numex) {
    int i = blockIdx.x * blockDim.x + threadIdx.x;
    if (i < B_ * M_) gtk[i] = fkey(-1.0f);
    if (i < B_)      sumfg[i] = 0;
    if (i == 0)      *numex = 0;
}

__global__ void k_overlaps(const float* __restrict__ gt, const float* __restrict__ anchors,
                           const float* __restrict__ img_info,
                           float* __restrict__ maxov, int* __restrict__ argm,
                           int* __restrict__ gtk) {
    __shared__ float sGT[M_ * 5 + 6];
    __shared__ int   sKey[M_];
    const int b = blockIdx.y, tid = threadIdx.x;
    const int lane = tid & 31;

    if (tid < 32) tdm_load_dwords(gt + (size_t)b * M_ * 5, sGT, M_ * 5, tid);
    if (tid < M_) sKey[tid] = fkey(-1.0f);
    __syncthreads();

    // T_ is an exact multiple of 256: every lane of every wave is active,
    // so wave32 butterfly reductions below are safe.
    const int t = blockIdx.x * blockDim.x + tid;
    {
        float imgH = img_info[0], imgW = img_info[1];
        float ax1, ay1, ax2, ay2; int xi, yi, ai;
        anchor_of(anchors, t, ax1, ay1, ax2, ay2, xi, yi, ai);
        bool inside = (ax1 >= 0.0f) && (ay1 >= 0.0f) && (ax2 < imgW) && (ay2 < imgH);
        float areaA = (ax2 - ax1 + 1.0f) * (ay2 - ay1 + 1.0f);
        const int KNEG1 = fkey(-1.0f);

        float mv = -2.0f; int am = 0;
        #pragma unroll 5
        for (int m = 0; m < M_; ++m) {
            float u = iou_one(ax1, ay1, ax2, ay2, areaA,
                              sGT[m * 5 + 0], sGT[m * 5 + 1], sGT[m * 5 + 2], sGT[m * 5 + 3]);
            float ov = inside ? u : -1.0f;
            if (ov > mv) { mv = ov; am = m; }
            // per-GT max over anchors: wave32 butterfly on monotonic int key,
            // then a single lane-0 LDS atomic (32x fewer LDS atomics).
            int key = inside ? fkey(u) : KNEG1;
            #pragma unroll
            for (int off = 16; off > 0; off >>= 1) {
                int o = __shfl_xor(key, off, 32);
                key = key > o ? key : o;
            }
            if (lane == 0) atomicMax(&sKey[m], key);
        }
        maxov[(size_t)b * T_ + t] = mv;
        argm [(size_t)b * T_ + t] = am;
    }
    __syncthreads();
    if (tid < M_) atomicMax(&gtk[b * M_ + tid], sKey[tid]);
}

__global__ void k_labels1(const float* __restrict__ gt, const float* __restrict__ anchors,
                          const float* __restrict__ img_info,
                          const float* __restrict__ maxov, const int* __restrict__ gtk,
                          float* __restrict__ lab1, int* __restrict__ sumfg) {
    __shared__ float sGT[M_ * 5 + 6];
    __shared__ float sMax[M_];
    const int b = blockIdx.y, tid = threadIdx.x;

    if (tid < 32) tdm_load_dwords(gt + (size_t)b * M_ * 5, sGT, M_ * 5, tid);
    if (tid < M_) {
        float f = kinv(gtk[b * M_ + tid]);
        if (f == 0.0f) f = 1e-5f;           // reference clamp
        sMax[tid] = f;
    }
    __syncthreads();

    const int t = blockIdx.x * blockDim.x + tid;
    {
        float imgH = img_info[0], imgW = img_info[1];
        float ax1, ay1, ax2, ay2; int xi, yi, ai;
        anchor_of(anchors, t, ax1, ay1, ax2, ay2, xi, yi, ai);
        bool inside = (ax1 >= 0.0f) && (ay1 >= 0.0f) && (ax2 < imgW) && (ay2 < imgH);

        float l = -1.0f;
        if (inside) {
            float areaA = (ax2 - ax1 + 1.0f) * (ay2 - ay1 + 1.0f);
            float mv = maxov[(size_t)b * T_ + t];
            if (mv < 0.3f) l = 0.0f;
            bool best = false;
            #pragma unroll 5
            for (int m = 0; m < M_; ++m) {
                float u = iou_one(ax1, ay1, ax2, ay2, areaA,
                                  sGT[m * 5 + 0], sGT[m * 5 + 1], sGT[m * 5 + 2], sGT[m * 5 + 3]);
                best = best || (u == sMax[m]);
            }
            if (best)       l = 1.0f;
            if (mv >= 0.7f) l = 1.0f;
            if (l == 1.0f) atomicAdd(&sumfg[b], 1);
        }
        lab1[(size_t)b * T_ + t] = l;
    }
}

// one block per (which in {fg,bg}, batch): histogram rank-selection with exact
// lex (value,index) cutoff inside the boundary bin (== stable argsort semantics)
__global__ void k_select(const float* __restrict__ lab1, const float* __restrict__ randu,
                         const int* __restrict__ sumfg, int* __restrict__ cut) {
    __shared__ int   hist[NBIN];
    __shared__ float vals[CAP_];
    __shared__ int   idxs[CAP_];
    __shared__ int   scnt, sbstar, skprime;
    __shared__ float scutv;
    __shared__ int   scuti;

    const int which = blockIdx.x, b = blockIdx.y, tid = threadIdx.x;
    const float want = (which == 0) ? 1.0f : 0.0f;
    const float* r = randu + (size_t)which * B_ * T_ + (size_t)b * T_;
    const float* L = lab1 + (size_t)b * T_;

    for (int i = tid; i < NBIN; i += blockDim.x) hist[i] = 0;
    if (tid == 0) { scnt = 0; scutv = 2.0f; scuti = 0; }
    __syncthreads();

    for (int t = tid; t < T_; t += blockDim.x) {
        if (L[t] == want) {
            int bin = (int)(r[t] * (float)NBIN); bin = bin > NBIN - 1 ? NBIN - 1 : bin;
            atomicAdd(&hist[bin], 1);
        }
    }
    __syncthreads();

    if (tid == 0) {
        int fa = sumfg[b]; if (fa > NUMFG) fa = NUMFG;
        int kt = (which == 0) ? NUMFG : (RPNB - fa);
        int cum = 0, bstar = NBIN, kp = 0;
        for (int bin = 0; bin < NBIN; ++bin) {
            int c = hist[bin];
            if (bstar == NBIN && (cum + c) >= kt) { bstar = bin; kp = kt - cum; }
            cum += c;
        }
        sbstar = bstar; skprime = kp;           // bstar==NBIN => total<kt => keep all
    }
    __syncthreads();

    const int bstar = sbstar, kp = skprime;
    if (bstar < NBIN) {
        for (int t = tid; t < T_; t += blockDim.x) {
            if (L[t] == want) {
                int bin = (int)(r[t] * (float)NBIN); bin = bin > NBIN - 1 ? NBIN - 1 : bin;
                if (bin == bstar) {
                    int p = atomicAdd(&scnt, 1);
                    if (p < CAP_) { vals[p] = r[t]; idxs[p] = t; }
                }
            }
        }
        __syncthreads();
        int n = scnt; if (n > CAP_) n = CAP_;
        if (tid < n) {
            float v = vals[tid]; int ix = idxs[tid]; int rank = 0;
            for (int l = 0; l < n; ++l) {
                float vl = vals[l]; int il = idxs[l];
                if (vl < v || (vl == v && il < ix)) rank++;
            }
            if (rank == kp) { scutv = v; scuti = ix; }   // unique (v,idx) => one writer
        }
        __syncthreads();
    }
    if (tid == 0) {
        int base = (which * B_ + b) * 3;
        cut[base + 0] = bstar;
        cut[base + 1] = __float_as_int(scutv);
        cut[base + 2] = scuti;
    }
}

// output-major: o = a*HW + y*W + x  =>  out[b*T + o] is lane-contiguous.
__global__ void k_final(const float* __restrict__ lab1, const float* __restrict__ randu,
                        const int* __restrict__ cut,
                        float* __restrict__ labF_tr, float* __restrict__ out_labels,
                        int* __restrict__ numex) {
    const int b = blockIdx.y;
    const int o = blockIdx.x * blockDim.x + threadIdx.x;
    if (o >= T_) return;
    const int HW = H_ * W_;
    int a = o / HW, r2 = o - a * HW;        // r2 = y*W + x
    int t = r2 * A_ + a;

    float l = lab1[(size_t)b * T_ + t];
    if (l == 1.0f || l == 0.0f) {
        int which = (l == 1.0f) ? 0 : 1;
        int base = (which * B_ + b) * 3;
        int   bstar = cut[base + 0];
        float cv    = __int_as_float(cut[base + 1]);
        int   ci    = cut[base + 2];
        float rv = randu[(size_t)which * B_ * T_ + (size_t)b * T_ + t];
        int bin = (int)(rv * (float)NBIN); bin = bin > NBIN - 1 ? NBIN - 1 : bin;
        bool keep = (bin < bstar) ||
                    (bin == bstar && (rv < cv || (rv == cv && t < ci)));
        if (!keep) l = -1.0f;
    }
    labF_tr   [(size_t)b * T_ + o] = l;     // transposed layout, coalesced
    out_labels[(size_t)b * T_ + o] = l;     // == ((b*A+a)*H+y)*W+x, coalesced
    if (b == B_ - 1 && l >= 0.0f) atomicAdd(numex, 1);
}

__global__ void k_targets(const float* __restrict__ gt, const float* __restrict__ anchors,
                          const float* __restrict__ img_info,
                          const int* __restrict__ argm, const float* __restrict__ labF_tr,
                          const int* __restrict__ numex,
                          float* __restrict__ out_t, float* __restrict__ out_iw,
                          float* __restrict__ out_ow) {
    const int b = blockIdx.y;
    const int o = blockIdx.x * blockDim.x + threadIdx.x;
    if (o >= T_) return;
    const int HW = H_ * W_;
    int a = o / HW, r2 = o - a * HW;
    int y = r2 / W_, x = r2 - y * W_;
    int t = r2 * A_ + a;

    float imgH = img_info[0], imgW = img_info[1];
    float sx = (float)x * 16.0f, sy = (float)y * 16.0f;
    float ax1 = anchors[a * 4 + 0] + sx;
    float ay1 = anchors[a * 4 + 1] + sy;
    float ax2 = anchors[a * 4 + 2] + sx;
    float ay2 = anchors[a * 4 + 3] + sy;
    bool inside = (ax1 >= 0.0f) && (ay1 >= 0.0f) && (ax2 < imgW) && (ay2 < imgH);

    int am = argm[(size_t)b * T_ + t];
    const float* g = gt + ((size_t)b * M_ + am) * 5;
    float ew = ax2 - ax1 + 1.0f, eh = ay2 - ay1 + 1.0f;
    float ex = ax1 + 0.5f * ew,  ey = ay1 + 0.5f * eh;
    float gw = g[2] - g[0] + 1.0f, gh = g[3] - g[1] + 1.0f;
    float gx = g[0] + 0.5f * gw,   gy = g[1] + 0.5f * gh;
    float tx = (gx - ex) / ew, ty = (gy - ey) / eh;
    float tw = logf(gw / ew);
    float th = logf(gh / eh);
    if (!inside) { tx = 0.0f; ty = 0.0f; tw = 0.0f; th = 0.0f; }

    float l  = labF_tr[(size_t)b * T_ + o];
    float iw = (l == 1.0f) ? 1.0f : 0.0f;
    float pw = 1.0f / (float)(*numex);
    float ow = (l == 1.0f || l == 0.0f) ? pw : 0.0f;

    size_t base = ((size_t)b * (A_ * 4) + a * 4) * HW + r2;   // coalesced per channel
    out_t [base + 0 * HW] = tx; out_t [base + 1 * HW] = ty;
    out_t [base + 2 * HW] = tw; out_t [base + 3 * HW] = th;
    out_iw[base + 0 * HW] = iw; out_iw[base + 1 * HW] = iw;
    out_iw[base + 2 * HW] = iw; out_iw[base + 3 * HW] = iw;
    out_ow[base + 0 * HW] = ow; out_ow[base + 1 * HW] = ow;
    out_ow[base + 2 * HW] = ow; out_ow[base + 3 * HW] = ow;
}

// ================= host side =================

extern "C" void kernel_launch(void* const* d_in, const int* in_sizes, int n_in,
                              void* d_out, int out_size, void* d_ws, size_t ws_size,
                              hipStream_t stream) {
    // inputs: 0 scores (unused), 1 gt_boxes (B,M,5), 2 img_info (B,2),
    //         3 rand_u (2,B,T), 4 anchors (9,4)
    const float* gt    = (const float*)d_in[1];
    const float* img   = (const float*)d_in[2];
    const float* randu = (const float*)d_in[3];
    const float* anch  = (const float*)d_in[4];

    const size_t BT = (size_t)B_ * T_;
    float* out_labels = (float*)d_out;                       // B*A*H*W
    float* out_targ   = out_labels + (size_t)B_ * A_ * H_ * W_;
    float* out_iw     = out_targ   + (size_t)B_ * A_ * 4 * H_ * W_;
    float* out_ow     = out_iw     + (size_t)B_ * A_ * 4 * H_ * W_;

    float* ws_maxov = (float*)d_ws;
    int*   ws_argm  = (int*)(ws_maxov + BT);
    float* ws_lab1  = (float*)(ws_argm + BT);
    float* ws_labF  = ws_lab1 + BT;            // transposed layout [b][a][y][x]
    int*   ws_gtk   = (int*)(ws_labF + BT);
    int*   ws_sumfg = ws_gtk + B_ * M_;
    int*   ws_numex = ws_sumfg + B_;
    int*   ws_cut   = ws_numex + 1;            // 2*B*3 ints

    dim3 blk(256);
    dim3 grid((T_ + 255) / 256, B_);

    k_init    <<<1, 1024, 0, stream>>>(ws_gtk, ws_sumfg, ws_numex);
    k_overlaps<<<grid, blk, 0, stream>>>(gt, anch, img, ws_maxov, ws_argm, ws_gtk);
    k_labels1 <<<grid, blk, 0, stream>>>(gt, anch, img, ws_maxov, ws_gtk, ws_lab1, ws_sumfg);
    k_select  <<<dim3(2, B_), 1024, 0, stream>>>(ws_lab1, randu, ws_sumfg, ws_cut);
    k_final   <<<grid, blk, 0, stream>>>(ws_lab1, randu, ws_cut, ws_labF, out_labels, ws_numex);
    k_targets <<<grid, blk, 0, stream>>>(gt, anch, img, ws_argm, ws_labF, ws_numex,
                                         out_targ, out_iw, out_ow);
}